// CMCModel_53128745452244
// MI455X (gfx1250) — compile-verified
//
#include <hip/hip_runtime.h>

// ---------------- problem constants (match reference) ----------------
#define T_ 64
#define B_ 32
#define M_ 8
#define A_ 16
#define H_ 128
#define F_ 152              // M + A + H
#define GI_ (M_ * H_)       // 1024
#define GR_ (H_ * H_)       // 16384
#define NGRP (M_ + H_ + 1)  // 137 groups of 128 columns (8 i, 128 R, 1 o)
#define WGSIZE 256
#define EPS_ 1e-5f

typedef __attribute__((ext_vector_type(2))) float v2f;
typedef __attribute__((ext_vector_type(8))) float v8f;
typedef __attribute__((ext_vector_type(4))) unsigned u32x4;
typedef __attribute__((ext_vector_type(8))) unsigned u32x8;

// ---------------- workspace layout (float offsets) ----------------
#define WS_C    0                    // c state              [B*H]
#define WS_F    (WS_C + B_ * H_)     // f = [xm,xa,c_n]      [B*F]
#define WS_MNEW (WS_F + B_ * F_)     // m_new accumulator    [B*H]
#define WS_O    (WS_MNEW + B_ * H_)  // o gate               [B*H]
#define WS_BAR  (WS_O + B_ * H_)     // 2 x u32 barrier state

__global__ void mclstm_init_bar(unsigned* bar) {
  if (threadIdx.x < 2) bar[threadIdx.x] = 0u;
}

__device__ __forceinline__ void grid_barrier(unsigned* cnt, unsigned* gen,
                                             unsigned nwg) {
  __syncthreads();
  if (threadIdx.x == 0) {
    unsigned g = __hip_atomic_load(gen, __ATOMIC_ACQUIRE, __HIP_MEMORY_SCOPE_AGENT);
    unsigned a = __hip_atomic_fetch_add(cnt, 1u, __ATOMIC_ACQ_REL, __HIP_MEMORY_SCOPE_AGENT);
    if (a == nwg - 1u) {
      __hip_atomic_store(cnt, 0u, __ATOMIC_RELAXED, __HIP_MEMORY_SCOPE_AGENT);
      __hip_atomic_fetch_add(gen, 1u, __ATOMIC_ACQ_REL, __HIP_MEMORY_SCOPE_AGENT);
    } else {
      while (__hip_atomic_load(gen, __ATOMIC_ACQUIRE, __HIP_MEMORY_SCOPE_AGENT) == g)
        __builtin_amdgcn_s_sleep(2);
    }
  }
  __syncthreads();
}

__global__ __launch_bounds__(WGSIZE)
void mclstm_persistent(const float* __restrict__ x_m,
                       const float* __restrict__ x_a,
                       const float* __restrict__ init_state,
                       const float* __restrict__ W_i, const float* __restrict__ b_i,
                       const float* __restrict__ W_r, const float* __restrict__ b_r,
                       const float* __restrict__ W_o, const float* __restrict__ b_o,
                       const float* __restrict__ fc_w, const float* __restrict__ fc_b,
                       float* __restrict__ out, float* __restrict__ ws) {
  __shared__ __align__(16) float fLDS[B_ * F_];  // 19.0 KB: f staged via TDM
  __shared__ __align__(16) float PL[B_ * H_];    // 16.0 KB: 32x128 preactivation tile
  __shared__ float red[8 * B_];
  __shared__ float red2[8 * B_];
  __shared__ float rowv[B_];

  float* c = ws + WS_C;
  float* fbuf = ws + WS_F;
  float* mnew = ws + WS_MNEW;
  float* obuf = ws + WS_O;
  unsigned* bar = (unsigned*)(ws + WS_BAR);

  const int tid = threadIdx.x;
  const int wg = blockIdx.x;
  const int lane = tid & 31;
  const int wave = tid >> 5;
  const int row = tid & 31;  // batch row for reductions (32 rows x 8 segs)
  const int seg = tid >> 5;  // 16-column segment

  // ----- this WG's 128-column output group -----
  const int g = wg;  // 0..136
  const float* Wp;
  const float* bp;
  int ncols, col0;
  if (g < M_)           { Wp = W_i; bp = b_i; ncols = GI_; col0 = g * H_; }
  else if (g < M_ + H_) { Wp = W_r; bp = b_r; ncols = GR_; col0 = (g - M_) * H_; }
  else                  { Wp = W_o; bp = b_o; ncols = H_;  col0 = 0; }

  const int arow = lane & 15;               // A-matrix row (batch) for this lane
  const int nloc0 = wave * 16;              // this wave's 16-column subtile
  const int ncol = col0 + nloc0 + arow;     // global W column for B operand
  const int kof = (lane < 16) ? 0 : 2;      // K offset per ISA f32 A/B layout
  const int prow0 = (lane < 16) ? 0 : 8;    // C/D row offset per ISA layout

  // ----- TDM descriptor: fbuf[B_ x F_] (global) -> fLDS (built once) -----
  // D# group 0: count=1 | lds_addr | 57-bit global_addr | type=2
  // D# group 1: data_size=4B, tensor 152x32, tile 152x32, dim0_stride=152
  u32x4 tdm_g0;
  u32x8 tdm_g1;
  {
    const unsigned long long ga = (unsigned long long)(const void*)fbuf;
    const unsigned lds_off = (unsigned)(unsigned long long)(const void*)&fLDS[0];
    tdm_g0.x = 1u;                                          // count=1, user D#
    tdm_g0.y = lds_off;                                     // LDS byte address
    tdm_g0.z = (unsigned)(ga & 0xFFFFFFFFull);              // global_addr lo
    tdm_g0.w = (unsigned)((ga >> 32) & 0x01FFFFFFull) | 0x80000000u;  // hi|type=2
    tdm_g1.s0 = 0x00020000u;                 // wg_mask=0, data_size=2 (4 bytes)
    tdm_g1.s1 = ((unsigned)F_ & 0xFFFFu) << 16;             // tensor_dim0[15:0]
    tdm_g1.s2 = ((unsigned)F_ >> 16) | (((unsigned)B_ & 0xFFFFu) << 16);
    tdm_g1.s3 = ((unsigned)B_ >> 16) | ((unsigned)F_ << 16);  // tile_dim0=F
    tdm_g1.s4 = (unsigned)B_;                // tile_dim1=B, tile_dim2=0
    tdm_g1.s5 = (unsigned)F_;                // tensor_dim0_stride lo32
    tdm_g1.s6 = 0u;
    tdm_g1.s7 = 0u;
  }

  // ----- t = 0 preparation (WG 0) -----
  if (wg == 0) {
    for (int idx = tid; idx < B_ * H_; idx += WGSIZE) {
      c[idx] = init_state[idx & (H_ - 1)];
      mnew[idx] = 0.0f;
    }
    __syncthreads();
    float cs = 0.0f;
    for (int j = 0; j < 16; ++j) cs += c[row * H_ + seg * 16 + j];
    red[seg * B_ + row] = cs;
    __syncthreads();
    if (tid < B_) {
      float s = 0.0f;
      for (int sg = 0; sg < 8; ++sg) s += red[sg * B_ + tid];
      rowv[tid] = 1.0f / (s + EPS_);
    }
    __syncthreads();
    for (int idx = tid; idx < B_ * F_; idx += WGSIZE) {
      int b = idx / F_, k = idx - b * F_;
      float v;
      if (k < M_)            v = x_m[b * M_ + k];
      else if (k < M_ + A_)  v = x_a[b * A_ + (k - M_)];
      else                   v = c[b * H_ + (k - M_ - A_)] * rowv[b];
      fbuf[idx] = v;
    }
  }
  __threadfence();
  grid_barrier(&bar[0], &bar[1], (unsigned)NGRP);

  for (int t = 0; t < T_; ++t) {
    // ---- stage f into LDS via Tensor Data Mover (one DMA per WG) ----
    if (wave == 0) {
      asm volatile("tensor_load_to_lds %0, %1"
                   :
                   : "s"(tdm_g0), "s"(tdm_g1)
                   : "memory");
      asm volatile("s_wait_tensorcnt 0x0" ::: "memory");
    }
    __syncthreads();

    // ---- fp32 WMMA GEMM: P[32x128] = f[32x152] @ W[:, col0:col0+128] ----
    v8f d0 = {};  // batches 0..15
    v8f d1 = {};  // batches 16..31 (shares B operand -> one pass over W/step)
    const float* wk = Wp + (size_t)kof * (size_t)ncols + (size_t)ncol;
    const size_t kstride = (size_t)4 * (size_t)ncols;
    const float* ap0 = &fLDS[arow * F_ + kof];
    const float* ap1 = ap0 + 16 * F_;
    for (int k0 = 0; k0 < F_; k0 += 4) {
      v2f bv;
      bv.x = wk[0];
      bv.y = wk[(size_t)ncols];
      __builtin_prefetch(wk + 2 * kstride, 0, 3);  // speculative, WGP scope
      v2f a0 = *(const v2f*)ap0;
      v2f a1 = *(const v2f*)ap1;
      d0 = __builtin_amdgcn_wmma_f32_16x16x4_f32(false, a0, false, bv,
                                                 (short)0, d0, false, false);
      d1 = __builtin_amdgcn_wmma_f32_16x16x4_f32(false, a1, false, bv,
                                                 (short)0, d1, false, false);
      wk += kstride;
      ap0 += 4;
      ap1 += 4;
    }

    // ---- add bias, spill 32x128 tile to LDS for cross-wave softmax ----
    const float biasv = bp[col0 + nloc0 + arow];
#pragma unroll
    for (int r = 0; r < 8; ++r) {
      PL[(prow0 + r) * H_ + nloc0 + arow] = d0[r] + biasv;
      PL[(16 + prow0 + r) * H_ + nloc0 + arow] = d1[r] + biasv;
    }
    __syncthreads();

    float* prow = &PL[row * H_ + seg * 16];
    if (g == M_ + H_) {
      // ---- output gate: sigmoid -> obuf ----
      for (int j = 0; j < 16; ++j) {
        float p = prow[j];
        obuf[row * H_ + seg * 16 + j] = 1.0f / (1.0f + __expf(-p));
      }
    } else {
      // ---- row softmax over 128 columns (max-subtracted) ----
      float mx = -3.4e38f;
      for (int j = 0; j < 16; ++j) mx = fmaxf(mx, prow[j]);
      red[seg * B_ + row] = mx;
      __syncthreads();
      if (tid < B_) {
        float m2 = red[tid];
        for (int sg = 1; sg < 8; ++sg) m2 = fmaxf(m2, red[sg * B_ + tid]);
        rowv[tid] = m2;
      }
      __syncthreads();
      const float rmax = rowv[row];
      float s = 0.0f;
      for (int j = 0; j < 16; ++j) {
        float e = __expf(prow[j] - rmax);
        prow[j] = e;
        s += e;
      }
      red2[seg * B_ + row] = s;
      __syncthreads();
      if (tid < B_) {
        float s2 = 0.0f;
        for (int sg = 0; sg < 8; ++sg) s2 += red2[sg * B_ + tid];
        rowv[tid] = s2;
      }
      __syncthreads();
      // ---- fold xm*i (g<M) or c*R (g>=M) into the shared m_new accumulator
      float wscale;
      if (g < M_) wscale = x_m[t * B_ * M_ + row * M_ + g];
      else        wscale = c[row * H_ + (g - M_)];
      const float fac = wscale / rowv[row];
      for (int j = 0; j < 16; ++j)
        atomicAdd(&mnew[row * H_ + seg * 16 + j], prow[j] * fac);
    }
    __threadfence();
    grid_barrier(&bar[0], &bar[1], (unsigned)NGRP);

    // ---- sequential cell update + logits + next f (WG 0 only) ----
    if (wg == 0) {
      float lsum = 0.0f, csum = 0.0f;
      const int base = row * H_ + seg * 16;
      for (int j = 0; j < 16; ++j) {
        const int h = seg * 16 + j;
        const float m = mnew[base + j];
        const float o = obuf[base + j];
        const float em = o * m;        // emitted mass
        const float cn = m - em;       // (1-o)*m_new
        c[base + j] = cn;
        mnew[base + j] = 0.0f;         // reset accumulator for next step
        lsum += em * fc_w[h];
        csum += cn;
      }
      red[seg * B_ + row] = lsum;
      red2[seg * B_ + row] = csum;
      __syncthreads();
      if (tid < B_) {
        float l = 0.0f, s = 0.0f;
        for (int sg = 0; sg < 8; ++sg) {
          l += red[sg * B_ + tid];
          s += red2[sg * B_ + tid];
        }
        out[t * B_ + tid] = l + fc_b[0];
        rowv[tid] = 1.0f / (s + EPS_);
      }
      __syncthreads();
      if (t + 1 < T_) {
        for (int idx = tid; idx < B_ * F_; idx += WGSIZE) {
          int b = idx / F_, k = idx - b * F_;
          float v;
          if (k < M_)            v = x_m[(t + 1) * B_ * M_ + b * M_ + k];
          else if (k < M_ + A_)  v = x_a[(t + 1) * B_ * A_ + b * A_ + (k - M_)];
          else                   v = c[b * H_ + (k - M_ - A_)] * rowv[b];
          fbuf[idx] = v;
        }
      } else {
        for (int idx = tid; idx < B_ * H_; idx += WGSIZE)
          out[T_ * B_ + idx] = c[idx];  // final cell state
      }
    }
    __threadfence();
    grid_barrier(&bar[0], &bar[1], (unsigned)NGRP);
  }
}

extern "C" void kernel_launch(void* const* d_in, const int* in_sizes, int n_in,
                              void* d_out, int out_size, void* d_ws,
                              size_t ws_size, hipStream_t stream) {
  (void)in_sizes; (void)n_in; (void)out_size; (void)ws_size;
  const float* x_m = (const float*)d_in[0];
  const float* x_a = (const float*)d_in[1];
  const float* init_state = (const float*)d_in[2];
  const float* W_i = (const float*)d_in[3];
  const float* b_i = (const float*)d_in[4];
  const float* W_r = (const float*)d_in[5];
  const float* b_r = (const float*)d_in[6];
  const float* W_o = (const float*)d_in[7];
  const float* b_o = (const float*)d_in[8];
  const float* fc_w = (const float*)d_in[9];
  const float* fc_b = (const float*)d_in[10];
  float* out = (float*)d_out;
  float* ws = (float*)d_ws;
  unsigned* bar = (unsigned*)(ws + WS_BAR);

  mclstm_init_bar<<<1, 64, 0, stream>>>(bar);
  mclstm_persistent<<<NGRP, WGSIZE, 0, stream>>>(
      x_m, x_a, init_state, W_i, b_i, W_r, b_r, W_o, b_o, fc_w, fc_b, out, ws);
}